// MotionTracking_8486855377009
// MI455X (gfx1250) — compile-verified
//
#include <hip/hip_runtime.h>
#include <hip/hip_bf16.h>

// Problem constants (match reference)
#define B_      8
#define NT_     256
#define WH_     21
#define P_      441
#define H_      1024
#define W_      1024
#define ITERS_  20

#define TRACKS_PER_BLOCK 8
#define LSTRIDE 23          // padded patch: rows -1..21 / cols -1..21 => 23x23, zero border

typedef float v2f __attribute__((ext_vector_type(2)));
typedef float v8f __attribute__((ext_vector_type(8)));

// Exact f32 all-lane (wave32) reduction:
//  - shfl_xor 1,2,4 -> every lane in each 8-group holds that group's sum s8(g), g = lane/8 in {0..3}
//  - one V_WMMA_F32_16X16X4_F32 with A = ones, B = (s8, 0):
//      D[m,n] = sum_k B[k,n] = b0(n) + b1(n) + b0(n+16) + b1(n+16)
//             = s8(n/8) + s8(n/8 + 2)            (cols 0-7: g0+g2, cols 8-15: g1+g3)
//    D vgpr0 gives each lane column (lane % 16)'s value.
//  - shfl_xor 8 pairs the two column halves -> every lane holds the full 32-lane sum.
__device__ __forceinline__ float wave_allreduce(float v) {
    v += __shfl_xor(v, 1, 32);
    v += __shfl_xor(v, 2, 32);
    v += __shfl_xor(v, 4, 32);
    v2f a = {1.0f, 1.0f};
    v2f b = {v, 0.0f};
    v8f c = {};
    v8f d = __builtin_amdgcn_wmma_f32_16x16x4_f32(
        /*neg_a=*/false, a, /*neg_b=*/false, b,
        /*c_mod=*/(short)0, c, /*reuse_a=*/false, /*reuse_b=*/false);
    float q = d[0];
    q += __shfl_xor(q, 8, 32);
    return q;
}

__device__ __forceinline__ float bilerp(const float* __restrict__ img, float x, float y) {
    float xf = floorf(x), yf = floorf(y);
    float x0 = fminf(fmaxf(xf,        0.f), (float)(W_ - 1));
    float x1 = fminf(fmaxf(xf + 1.f,  0.f), (float)(W_ - 1));
    float y0 = fminf(fmaxf(yf,        0.f), (float)(H_ - 1));
    float y1 = fminf(fmaxf(yf + 1.f,  0.f), (float)(H_ - 1));
    int x0i = (int)x0, x1i = (int)x1, y0i = (int)y0, y1i = (int)y1;
    float wa = (y1 - y) * (x1 - x);
    float wb = (y1 - y) * (x - x0);
    float wc = (y - y0) * (x1 - x);
    float wd = (y - y0) * (x - x0);
    const float* r0 = img + (size_t)y0i * W_;
    const float* r1 = img + (size_t)y1i * W_;
    return wa * r0[x0i] + wb * r0[x1i] + wc * r1[x0i] + wd * r1[x1i];
}

__global__ __launch_bounds__(256)
void MotionTracking_kernel(const float* __restrict__ track_locs,
                           const float* __restrict__ imgs,
                           float* __restrict__ out) {
    __shared__ float lds[TRACKS_PER_BLOCK][LSTRIDE * 23];

    const int lane  = threadIdx.x & 31;
    const int wave  = threadIdx.x >> 5;
    const int track = blockIdx.x * TRACKS_PER_BLOCK + wave;
    const int b     = track / NT_;
    const int nt    = track % NT_;
    float* patch = lds[wave];

    // Zero LDS patch (border stays zero forever -> branchless SAME-padded Sobel)
    for (int i = lane; i < LSTRIDE * 23; i += 32) patch[i] = 0.f;

    // init = track_locs / SCALE, SCALE = WH/min(W,H) = 21/1024 (exact in binary)
    const float SCALE = (float)WH_ / (float)(W_ < H_ ? W_ : H_);
    float cx = track_locs[(b * NT_ + nt) * 2 + 0] / SCALE;
    float cy = track_locs[(b * NT_ + nt) * 2 + 1] / SCALE;

    const float* f0 = imgs + (size_t)(b * 2 + 0) * H_ * W_;
    const float* f1 = imgs + (size_t)(b * 2 + 1) * H_ * W_;

    // Warm the L2/L0 path for the initial ~22-row patch region (global_prefetch_b8)
    {
        float px = (cx - 1.0f) * (WH_ * 0.5f);
        float py = (cy - 1.0f) * (WH_ * 0.5f);
        int xi = min(max((int)px, 0), W_ - 1);
        int yi = min(max((int)py + lane, 0), H_ - 1);
        if (lane < 22) {
            __builtin_prefetch(f0 + (size_t)yi * W_ + xi, 0, 3);
            __builtin_prefetch(f1 + (size_t)yi * W_ + xi, 0, 3);
        }
    }

    float firstv[14];
    float secondv[14];

    // ---- initial sampling: frame0 (patch + regs) and frame1 (fixed "second") ----
    #pragma unroll
    for (int t = 0; t < 14; ++t) {
        int k = lane + 32 * t;
        float v0 = 0.f, v1 = 0.f;
        if (k < P_) {
            int i = k / WH_, j = k % WH_;
            float gx = -1.0f + 0.1f * (float)j;     // linspace(-1,1,21)
            float gy = -1.0f + 0.1f * (float)i;
            float x = (gx + cx) * (WH_ * 0.5f);
            float y = (gy + cy) * (WH_ * 0.5f);
            v0 = bilerp(f0, x, y);
            v1 = bilerp(f1, x, y);
            patch[(i + 1) * LSTRIDE + (j + 1)] = v0;
        }
        firstv[t]  = v0;
        secondv[t] = v1;
    }
    __syncthreads();

    // ---- 20 Lucas-Kanade iterations ----
    for (int it = 0; it < ITERS_; ++it) {
        float sxx = 0.f, syy = 0.f, sxy = 0.f, sxt = 0.f, syt = 0.f;
        #pragma unroll
        for (int t = 0; t < 14; ++t) {
            int k = lane + 32 * t;
            if (k < P_) {
                int i = k / WH_ + 1, j = k % WH_ + 1;
                const float* c = &patch[i * LSTRIDE + j];
                float pmm = c[-LSTRIDE - 1], pm0 = c[-LSTRIDE], pmp = c[-LSTRIDE + 1];
                float p0m = c[-1],                               p0p = c[1];
                float ppm = c[ LSTRIDE - 1], pp0 = c[ LSTRIDE], ppp = c[ LSTRIDE + 1];
                float Ix = (pmp + 2.f * p0p + ppp) - (pmm + 2.f * p0m + ppm);
                float Iy = (ppm + 2.f * pp0 + ppp) - (pmm + 2.f * pm0 + pmp);
                float It = secondv[t] - firstv[t];
                sxx += Ix * Ix; syy += Iy * Iy; sxy += Ix * Iy;
                sxt += Ix * It; syt += Iy * It;
            }
        }
        float Ix2 = wave_allreduce(sxx);
        float Iy2 = wave_allreduce(syy);
        float Ixy = wave_allreduce(sxy);
        float Ixt = wave_allreduce(sxt);
        float Iyt = wave_allreduce(syt);

        float det_inv = 1.0f / (Ix2 * Iy2 - Ixy * Ixy);
        float Vx = det_inv * (Ixy * Iyt - Iy2 * Ixt);
        float Vy = det_inv * (Ixy * Ixt - Ix2 * Iyt);
        cx -= Vx;
        cy -= Vy;

        __syncthreads();   // everyone done reading patch before overwrite
        #pragma unroll
        for (int t = 0; t < 14; ++t) {
            int k = lane + 32 * t;
            if (k < P_) {
                int i = k / WH_, j = k % WH_;
                float gx = -1.0f + 0.1f * (float)j;
                float gy = -1.0f + 0.1f * (float)i;
                float x = (gx + cx) * (WH_ * 0.5f);
                float y = (gy + cy) * (WH_ * 0.5f);
                float v0 = bilerp(f0, x, y);
                firstv[t] = v0;
                patch[(i + 1) * LSTRIDE + (j + 1)] = v0;
            }
        }
        __syncthreads();
    }

    // ---- output: (B, 2, NT, WH, WH, 1): ch0 = final frame0 patch, ch1 = fixed second ----
    float* out0 = out + ((size_t)(b * 2 + 0) * NT_ + nt) * P_;
    float* out1 = out + ((size_t)(b * 2 + 1) * NT_ + nt) * P_;
    #pragma unroll
    for (int t = 0; t < 14; ++t) {
        int k = lane + 32 * t;
        if (k < P_) {
            out0[k] = firstv[t];
            out1[k] = secondv[t];
        }
    }
}

extern "C" void kernel_launch(void* const* d_in, const int* in_sizes, int n_in,
                              void* d_out, int out_size, void* d_ws, size_t ws_size,
                              hipStream_t stream) {
    const float* track_locs = (const float*)d_in[0];   // (B, NT*2) f32
    const float* imgs       = (const float*)d_in[1];   // (B, 2, H, W, 1) f32
    float* out              = (float*)d_out;           // (B, 2, NT, WH, WH, 1) f32

    dim3 grid((B_ * NT_) / TRACKS_PER_BLOCK);          // 256 blocks
    dim3 block(256);                                   // 8 wave32 = 8 tracks/block
    hipLaunchKernelGGL(MotionTracking_kernel, grid, block, 0, stream,
                       track_locs, imgs, out);
}